// MaskCapsule_30966714204524
// MI455X (gfx1250) — compile-verified
//
#include <hip/hip_runtime.h>
#include <math.h>

// Problem constants (from setup_inputs): batch=128, HW=4, B=512, cp=1024, cn=5
#define BQ    512
#define CP    1024
#define CN    5
#define NWAVE 32

// d_out layout (flat, return-tuple order):
//   [0,8192)       mask_output   (512,16)
//   [8192,10752)   out_act       (512,5)
//   [10752,18944)  mask_output2  (512,16)
//   [18944,59904)  mean_v        (128,4,5,16) == (b*5+n)*16+k
#define OFF_ACT   8192
#define OFF_MASK2 10752
#define OFF_MEANV 18944

typedef float v2f  __attribute__((ext_vector_type(2)));
typedef float v8f  __attribute__((ext_vector_type(8)));
typedef unsigned int u32x4 __attribute__((ext_vector_type(4)));
typedef int   i32x4 __attribute__((ext_vector_type(4)));
typedef int   i32x8 __attribute__((ext_vector_type(8)));

#define TWO_PI_F 6.283185307179586f

// DPP16 control codes
#define DPP_QUAD_XOR1   0xB1   // quad_perm [1,0,3,2]
#define DPP_QUAD_XOR2   0x4E   // quad_perm [2,3,0,1]
#define DPP_HALF_MIRROR 0x141  // lane ^ 7 within rows of 8
#define DPP_MIRROR      0x140  // lane ^ 15 within rows of 16

// All patterns fetch valid lanes under full EXEC, so old=0 + bound_ctrl=true is
// safe and lets the backend emit a single v_mov_b32_dpp (no extra v_mov).
template <int CTRL>
__device__ __forceinline__ float dpp_add(float v) {
    const int s = __builtin_amdgcn_update_dpp(
        0, __float_as_int(v), CTRL, 0xF, 0xF, true);
    return v + __int_as_float(s);
}

// Butterfly sum over 16-lane k-groups, pure DPP (no LDS, no waits).
// half_mirror/mirror are xor4/xor8-equivalent once values are quad/oct-uniform.
__device__ __forceinline__ float ksum16(float v) {
    v = dpp_add<DPP_QUAD_XOR1>(v);
    v = dpp_add<DPP_QUAD_XOR2>(v);
    v = dpp_add<DPP_HALF_MIRROR>(v);
    v = dpp_add<DPP_MIRROR>(v);
    return v;
}

__global__ __launch_bounds__(1024)
void capsule_em_kernel(const float* __restrict__ pose,     // (512,1024,4,4)
                       const float* __restrict__ act,      // (512,1024)
                       const float* __restrict__ ground,   // (128,5)
                       const float* __restrict__ ground2,  // (128,5)
                       const float* __restrict__ W,        // (5,1024,4,4)
                       const float* __restrict__ Bv,       // (5,16)
                       const float* __restrict__ Ba,       // (5,)
                       const int*   __restrict__ itemp,    // scalar
                       float* __restrict__ out)
{
    // ---- LDS (~110 KB of 320 KB WGP budget) ----
    __shared__ float poseS[CP * 16];        // 64 KB : TDM destination
    __shared__ float rS[CN][CP];            // 20 KB : routing * activation
    __shared__ float actS[CP];              //  4 KB
    __shared__ float SW[CN][NWAVE][16];     // 10 KB : per-wave S partials
    __shared__ float QW[CN][NWAVE][16];     // 10 KB : per-wave Q partials
    __shared__ float rsumW[CN][NWAVE];
    __shared__ float meanS[CN][16];
    __shared__ float aS[CN][16];            // 1/(2*var+1e-5)
    __shared__ float varS[CN][16];
    __shared__ float costS[CN][16];
    __shared__ float outactS[CN];
    __shared__ float dinvS[CN];             // 1/(denom+1e-4)

    const int tid   = threadIdx.x;
    const int b     = blockIdx.x;
    const int lane  = tid & 31;
    const int wid   = tid >> 5;
    const int kk    = lane & 15;   // k index (0..15) — WMMA N / lane dim
    const int h     = lane >> 4;   // lane half  — WMMA K-slot high bit
    const int wbase = wid << 5;    // this wave owns p in [wbase, wbase+32)

    // ---- stage activation, init routing*act = act/cn ----
    {
        float a = act[(size_t)b * CP + tid];
        actS[tid] = a;
        const float r0 = a * (1.0f / (float)CN);
#pragma unroll
        for (int n = 0; n < CN; n++) rS[n][tid] = r0;
    }

    // ---- TDM: async DMA pose[b] (1024x16 f32 = 64 KB, contiguous) -> LDS ----
    if (tid == 0) {
        const unsigned long long gaddr =
            (unsigned long long)(const void*)(pose + (size_t)b * CP * 16);
        const unsigned int ldsOff = (unsigned int)(unsigned long long)(void*)poseS;
        // D# group0: count=1, lds_addr, global_addr[56:0], type=2
        u32x4 g0 = { 1u,
                     ldsOff,
                     (unsigned int)gaddr,
                     (unsigned int)(((gaddr >> 32) & 0x1FFFFFFu) | (2u << 30)) };
        // D# group1: wg_mask=0, data_size=2(4B), dims: tensor_dim0=tile_dim0=16384
        i32x8 g1 = { (int)(2u << 16),        // data_size = 4 bytes
                     (int)(16384u << 16),    // tensor_dim0[15:0] in [31:16]
                     (int)(1u << 16),        // tensor_dim0 hi=0 | tensor_dim1=1
                     (int)(16384u << 16),    // tile_dim0 = 16384
                     1,                      // tile_dim1=1, tile_dim2=0
                     16384,                  // tensor_dim0_stride
                     (int)(16384u << 16),    // tensor_dim1_stride[15:0]
                     0 };
        i32x4 g2 = { 0, 0, 0, 0 };
        i32x4 g3 = { 0, 0, 0, 0 };
#if defined(__clang_major__) && __clang_major__ >= 23
        i32x8 g4 = { 0, 0, 0, 0, 0, 0, 0, 0 };
        __builtin_amdgcn_tensor_load_to_lds(g0, g1, g2, g3, g4, 0);
#else
        __builtin_amdgcn_tensor_load_to_lds(g0, g1, g2, g3, 0);
#endif
        __builtin_amdgcn_s_wait_tensorcnt(0);
    }
    __syncthreads();

    // ---- vote compute: lane owns k=kk; K-slots (h,v) own p = wbase+4c+2h+v ----
    // vote[n][2c+v] = vote_global[b, n, p(c,h,v), kk]; this IS the WMMA B layout.
    float vote[CN][16];
    {
#pragma unroll
        for (int n = 0; n < CN; n++)
            __builtin_prefetch(W + ((size_t)(n * CP + wbase) * 16), 0, 0);

        const int i4 = (kk >> 2) * 4;  // W row base: 4*i
        const int kc = kk & 3;         // pose column
#pragma unroll
        for (int c = 0; c < 8; c++) {
#pragma unroll
            for (int v = 0; v < 2; v++) {
                const int p  = wbase + 4 * c + 2 * h + v;
                const float p0 = poseS[p * 16 + 0  + kc];
                const float p1 = poseS[p * 16 + 4  + kc];
                const float p2 = poseS[p * 16 + 8  + kc];
                const float p3 = poseS[p * 16 + 12 + kc];
#pragma unroll
                for (int n = 0; n < CN; n++) {
                    const float4 w4 =
                        *(const float4*)(W + ((size_t)(n * CP + p) * 16 + i4));
                    vote[n][2 * c + v] = w4.x * p0 + w4.y * p1 + w4.z * p2 + w4.w * p3;
                }
            }
        }
    }

    const float it0 = (float)itemp[0];

#pragma unroll 1
    for (int iter = 0; iter < 3; iter++) {
        const float inv_t = it0 + (float)iter;

        // ---- S = Σ_p r·v and Q = Σ_p r·v² via V_WMMA_F32_16X16X4_F32 ----
        // A[*,K] = r[n, p(K)] (rows replicated), B[K,N=k] = vote (native layout).
        // S and Q accumulated in separate passes to keep one v8f live (no spills).
#pragma unroll
        for (int n = 0; n < CN; n++) {
            v8f cs = { 0.f, 0.f, 0.f, 0.f, 0.f, 0.f, 0.f, 0.f };
#pragma unroll
            for (int c = 0; c < 8; c++) {
                const int pb = wbase + 4 * c + 2 * h;
                v2f A;  A.x  = rS[n][pb];      A.y  = rS[n][pb + 1];
                v2f Bm; Bm.x = vote[n][2 * c]; Bm.y = vote[n][2 * c + 1];
                cs = __builtin_amdgcn_wmma_f32_16x16x4_f32(
                         false, A, false, Bm, (short)0, cs, false, false);
            }
            if (lane < 16) SW[n][wid][lane] = cs[0];

            v8f cq = { 0.f, 0.f, 0.f, 0.f, 0.f, 0.f, 0.f, 0.f };
#pragma unroll
            for (int c = 0; c < 8; c++) {
                const int pb = wbase + 4 * c + 2 * h;
                v2f A;  A.x  = rS[n][pb];      A.y  = rS[n][pb + 1];
                const float b0 = vote[n][2 * c], b1 = vote[n][2 * c + 1];
                v2f Bq; Bq.x = b0 * b0;        Bq.y = b1 * b1;
                cq = __builtin_amdgcn_wmma_f32_16x16x4_f32(
                         false, A, false, Bq, (short)0, cq, false, false);
            }
            if (lane < 16) QW[n][wid][lane] = cq[0];

            // rsum partial for this wave's 32 p's: DPP butterfly + one cross-row
            float rv = rS[n][wbase + lane];
            rv = ksum16(rv);
            rv += __shfl_xor(rv, 16);
            if (lane == 0) rsumW[n][wid] = rv;
        }
        __syncthreads();

        // ---- per-(n,k) stats: deterministic fixed-order cross-wave combine ----
        if (tid < CN * 16) {
            const int n = tid >> 4, k = tid & 15;
            float rs = 0.f, S = 0.f, Q = 0.f;
#pragma unroll 1
            for (int w = 0; w < NWAVE; w++) {
                rs += rsumW[n][w];  S += SW[n][w][k];  Q += QW[n][w][k];
            }
            const float m = S / (rs + 1e-5f);
            float var = (Q - 2.f * m * S + m * m * rs) / (rs + 1e-3f);
            var = fmaxf(var, 0.f);
            meanS[n][k] = m;
            varS[n][k]  = var;
            aS[n][k]    = 1.f / (2.f * var + 1e-5f);
            costS[n][k] = (Bv[n * 16 + k] + logf(sqrtf(var + 1e-6f) + 1e-5f)) * rs;
        }
        __syncthreads();

        // ---- per-n stats: out_act (sigmoid over z-scored cost), denom ----
        if (tid < CN) {
            float sc[CN]; float scm = 0.f;
#pragma unroll
            for (int i = 0; i < CN; i++) {
                float s = 0.f;
#pragma unroll
                for (int k = 0; k < 16; k++) s += costS[i][k];
                sc[i] = s; scm += s;
            }
            scm *= (1.0f / (float)CN);
            float dev = 0.f;
#pragma unroll
            for (int i = 0; i < CN; i++) { const float d = sc[i] - scm; dev += d * d; }
            const float stdv = sqrtf(dev * (1.0f / (float)CN) + 1e-4f);
            float vsum = 0.f;
#pragma unroll
            for (int k = 0; k < 16; k++) vsum += varS[tid][k];
            const float x = inv_t * (Ba[tid] + (scm - sc[tid]) / (stdv + 1e-5f));
            outactS[tid] = 1.f / (1.f + __expf(-x));
            dinvS[tid]   = 1.f / (sqrtf(TWO_PI_F * vsum + 1e-5f) + 1e-4f);
        }
        __syncthreads();

        // ---- prob + routing update (results unused after last iteration) ----
        if (iter < 2) {
            float mreg[CN], areg[CN], oact[CN], dv[CN];
#pragma unroll
            for (int n = 0; n < CN; n++) {
                mreg[n] = meanS[n][kk]; areg[n] = aS[n][kk];
                oact[n] = outactS[n];   dv[n]   = dinvS[n];
            }
#pragma unroll
            for (int j = 0; j < 16; j++) {
                float ap[CN]; float sum = 0.f;
#pragma unroll
                for (int n = 0; n < CN; n++) {
                    const float t = vote[n][j] - mreg[n];
                    const float e = ksum16(t * t * areg[n]); // Σ_k via DPP
                    const float pr = __expf(-e) * dv[n];
                    ap[n] = oact[n] * pr;
                    sum += ap[n];
                }
                const int p = wbase + 4 * (j >> 1) + 2 * h + (j & 1);
                const float scale = actS[p] / (sum + 1e-5f); // fold next iter's *act
                if (kk == j) {
#pragma unroll
                    for (int n = 0; n < CN; n++) rS[n][p] = ap[n] * scale;
                }
            }
        }
        __syncthreads();
    }

    // ---- outputs (mean & out_act from last iteration) ----
    if (tid < CN * 16) {
        const int n = tid >> 4, k = tid & 15;
        out[OFF_MEANV + (b * CN + n) * 16 + k] = meanS[n][k];
    }
    if (tid < CN) out[OFF_ACT + b * CN + tid] = outactS[tid];
    if (tid < 16) {
        const int batch = b >> 2;
        float m1 = 0.f, m2 = 0.f;
#pragma unroll
        for (int n = 0; n < CN; n++) {
            m1 += meanS[n][tid] * ground [batch * CN + n];
            m2 += meanS[n][tid] * ground2[batch * CN + n];
        }
        out[b * 16 + tid]             = m1;
        out[OFF_MASK2 + b * 16 + tid] = m2;
    }
}

extern "C" void kernel_launch(void* const* d_in, const int* in_sizes, int n_in,
                              void* d_out, int out_size, void* d_ws, size_t ws_size,
                              hipStream_t stream) {
    (void)in_sizes; (void)n_in; (void)d_ws; (void)ws_size; (void)out_size;
    capsule_em_kernel<<<BQ, 1024, 0, stream>>>(
        (const float*)d_in[0],   // inputPose
        (const float*)d_in[1],   // inputActivation
        (const float*)d_in[2],   // ground
        (const float*)d_in[3],   // ground2
        (const float*)d_in[4],   // weightMatrix
        (const float*)d_in[5],   // Bv
        (const float*)d_in[6],   // Ba
        (const int*)  d_in[9],   // inverse_temperature
        (float*)d_out);
}